// LKA3d_67284957659556
// MI455X (gfx1250) — compile-verified
//
#include <hip/hip_runtime.h>
#include <hip/hip_bf16.h>

#define CDIM 64
#define PVOL (48*48*48)   // 110592 spatial positions

typedef __attribute__((ext_vector_type(2))) float v2f;
typedef __attribute__((ext_vector_type(8))) float v8f;

// ---------------------------------------------------------------------------
// Kernel A: depthwise 5x5x5 conv, pad 2. LDS-tiled: 8^3 outputs, 12^3 halo.
// ---------------------------------------------------------------------------
__global__ __launch_bounds__(512) void dwconv5_kernel(
    const float* __restrict__ x, const float* __restrict__ w0,
    const float* __restrict__ b0, float* __restrict__ out)
{
  __shared__ float tile[12*12*12];
  const int t  = blockIdx.x;              // 0..215 (6x6x6 tiles)
  const int tx = t % 6, ty = (t / 6) % 6, tz = t / 36;
  const int c  = blockIdx.y;
  const int nb = blockIdx.z;
  const float* xin = x + (size_t)(nb*CDIM + c) * PVOL;

  // cooperative halo load (1728 elems / 512 threads)
  for (int i = threadIdx.x; i < 12*12*12; i += 512) {
    int lx = i % 12, ly = (i/12) % 12, lz = i / 144;
    int gx = tx*8 + lx - 2, gy = ty*8 + ly - 2, gz = tz*8 + lz - 2;
    float v = 0.f;
    if (gx >= 0 && gx < 48 && gy >= 0 && gy < 48 && gz >= 0 && gz < 48)
      v = xin[(gz*48 + gy)*48 + gx];
    tile[i] = v;
  }
  __syncthreads();

  const int ox = threadIdx.x % 8, oy = (threadIdx.x/8) % 8, oz = threadIdx.x/64;
  const float* wc = w0 + c*125;           // uniform per block -> scalar loads
  float acc = b0[c];
  #pragma unroll
  for (int kz = 0; kz < 5; ++kz)
    #pragma unroll
    for (int ky = 0; ky < 5; ++ky)
      #pragma unroll
      for (int kx = 0; kx < 5; ++kx)
        acc = fmaf(tile[((oz+kz)*12 + (oy+ky))*12 + (ox+kx)],
                   wc[(kz*5+ky)*5+kx], acc);

  int gx = tx*8+ox, gy = ty*8+oy, gz = tz*8+oz;
  out[(size_t)(nb*CDIM+c)*PVOL + (gz*48+gy)*48 + gx] = acc;
}

// ---------------------------------------------------------------------------
// Kernel B: depthwise 7x7x7 conv, dilation 3, pad 9.  Input tensor (56.6 MB)
// fits in the 192 MB L2, so direct global loads are L2-resident.
// ---------------------------------------------------------------------------
__global__ __launch_bounds__(256) void dwconv7d3_kernel(
    const float* __restrict__ in, const float* __restrict__ ws,
    const float* __restrict__ bs, float* __restrict__ out)
{
  const int idx = blockIdx.x * 256 + threadIdx.x;   // exact cover of 2*64*PVOL
  const int wx = idx % 48;
  const int hy = (idx / 48) % 48;
  const int dz = (idx / 2304) % 48;
  const int c  = (idx / PVOL) % CDIM;
  const int nb = idx / (PVOL*CDIM);
  const float* xin = in + (size_t)(nb*CDIM + c) * PVOL;
  const float* wc  = ws + c*343;

  float acc = bs[c];
  for (int kz = 0; kz < 7; ++kz) {
    int z = dz + (kz-3)*3;
    if (z < 0 || z >= 48) continue;
    for (int ky = 0; ky < 7; ++ky) {
      int y = hy + (ky-3)*3;
      if (y < 0 || y >= 48) continue;
      const float* row = xin + (z*48 + y)*48;
      const float* wr  = wc + (kz*7+ky)*7;
      #pragma unroll
      for (int kx = 0; kx < 7; ++kx) {
        int xx = wx + (kx-3)*3;
        if (xx >= 0 && xx < 48)
          acc = fmaf(row[xx], wr[kx], acc);
      }
    }
  }
  out[idx] = acc;
}

// ---------------------------------------------------------------------------
// Kernel C: pointwise 1x1x1 conv as f32 GEMM on the WMMA pipe
//   out[co,p] = ( sum_ci W[co,ci] * attn[ci,p] + b1[co] ) * x[co,p]
// One wave -> one 16x16 (co x pos) tile, K=64 via 16x V_WMMA_F32_16X16X4_F32.
// Block = 256 threads = 8 waves = 4 co-tiles x 2 pos-tiles (32 positions).
// ---------------------------------------------------------------------------
__global__ __launch_bounds__(256) void pw_wmma_gate_kernel(
    const float* __restrict__ attn, const float* __restrict__ w1,
    const float* __restrict__ b1, const float* __restrict__ x,
    float* __restrict__ out)
{
  const int lane    = threadIdx.x & 31;
  const int wv      = threadIdx.x >> 5;        // wave 0..7
  const int co_base = (wv & 3) * 16;
  const int pos_base = blockIdx.x * 32 + (wv >> 2) * 16;
  const int nb      = blockIdx.y;

  const int m   = lane & 15;                   // A row / B,C column
  const int hi  = lane >> 4;                   // lane half
  const int klo = hi * 2;                      // K sub-offset within fragment

  const float* A2 = attn + (size_t)nb * CDIM * PVOL;

  // C/D layout: vgpr v, lane L -> row = v + 8*(L>=16). Fold bias in here.
  v8f c;
  #pragma unroll
  for (int v = 0; v < 8; ++v) c[v] = b1[co_base + v + 8*hi];

  #pragma unroll
  for (int ks = 0; ks < 16; ++ks) {
    const int k0 = ks * 4;
    // A fragment (16x4, W is row-major [co][ci]): lane holds row m,
    // vgpr j holds k = k0 + klo + j  -> contiguous pair, single b64 load.
    const float* wp = w1 + (co_base + m)*CDIM + (k0 + klo);
    v2f a; a[0] = wp[0]; a[1] = wp[1];
    // B fragment (4x16): lane holds column (position) m, vgpr j holds
    // k = k0 + klo + j -> stride PVOL between the two elements.
    const float* bp = A2 + (size_t)(k0 + klo) * PVOL + (pos_base + m);
    v2f bf; bf[0] = bp[0]; bf[1] = bp[PVOL];

    c = __builtin_amdgcn_wmma_f32_16x16x4_f32(
        /*neg_a=*/false, a, /*neg_b=*/false, bf,
        /*c_mod=*/(short)0, c, /*reuse_a=*/false, /*reuse_b=*/false);
  }

  // Fused residual gate + store
  const float* xb = x   + (size_t)nb * CDIM * PVOL;
  float*       ob = out + (size_t)nb * CDIM * PVOL;
  #pragma unroll
  for (int v = 0; v < 8; ++v) {
    int co = co_base + v + 8*hi;
    size_t o = (size_t)co * PVOL + (pos_base + m);
    ob[o] = c[v] * xb[o];
  }
}

// ---------------------------------------------------------------------------
extern "C" void kernel_launch(void* const* d_in, const int* in_sizes, int n_in,
                              void* d_out, int out_size, void* d_ws, size_t ws_size,
                              hipStream_t stream) {
  const float* x  = (const float*)d_in[0];
  const float* w0 = (const float*)d_in[1];
  const float* b0 = (const float*)d_in[2];
  const float* ws = (const float*)d_in[3];
  const float* bs = (const float*)d_in[4];
  const float* w1 = (const float*)d_in[5];
  const float* b1 = (const float*)d_in[6];
  float* out = (float*)d_out;

  const size_t n_elem = (size_t)2 * CDIM * PVOL;   // 14,155,776
  float* attn1 = (float*)d_ws;
  float* attn2 = attn1 + n_elem;

  // conv0: depthwise 5^3
  dim3 gA(216, CDIM, 2);
  dwconv5_kernel<<<gA, 512, 0, stream>>>(x, w0, b0, attn1);

  // conv_spatial: depthwise 7^3, dilation 3
  dwconv7d3_kernel<<<(unsigned)(n_elem / 256), 256, 0, stream>>>(attn1, ws, bs, attn2);

  // conv1 (pointwise GEMM on WMMA) + bias + residual gate
  dim3 gC(PVOL / 32, 2, 1);
  pw_wmma_gate_kernel<<<gC, 256, 0, stream>>>(attn2, w1, b1, x, out);
}